// Conv_39273180955618
// MI455X (gfx1250) — compile-verified
//
#include <hip/hip_runtime.h>
#include <stdint.h>

typedef float v2f __attribute__((ext_vector_type(2)));
typedef float v4f __attribute__((ext_vector_type(4)));
typedef float v8f __attribute__((ext_vector_type(8)));
typedef int   v4i __attribute__((ext_vector_type(4)));

#define PLANES   1024      // 16 * 64
#define W        224
#define H        224
#define ROWS     32        // output rows per block
#define LROWS    34        // input rows per block (ROWS + 2 halo)
#define BANDS    7         // 224 / 32
#define STRD     232       // LDS floats per row: [0..3] left pad, [4..227] data, [228] right pad, [229..231] align pad

// ---------- CDNA5 async global->LDS helpers ----------
__device__ __forceinline__ void async_load_f4(const float* gsrc, float* ldst) {
#if __has_builtin(__builtin_amdgcn_global_load_async_to_lds_b128)
  typedef __attribute__((address_space(1))) v4i* gv4i_p;
  typedef __attribute__((address_space(3))) v4i* lv4i_p;
  __builtin_amdgcn_global_load_async_to_lds_b128(
      (gv4i_p)(uintptr_t)gsrc,
      (lv4i_p)(uint32_t)(uintptr_t)ldst,
      0, 0);
#else
  unsigned lo = (uint32_t)(uintptr_t)ldst;
  asm volatile("global_load_async_to_lds_b128 %0, %1, off"
               :: "v"(lo), "v"(gsrc) : "memory");
#endif
}

__device__ __forceinline__ void wait_async0() {
#if __has_builtin(__builtin_amdgcn_s_wait_asynccnt)
  __builtin_amdgcn_s_wait_asynccnt(0);
#else
  asm volatile("s_wait_asynccnt 0" ::: "memory");
#endif
}

// ---------- kernel 1: Keff[9] + bias_total -> ws[0..9] ----------
__global__ __launch_bounds__(256) void conv_reduce(const float* __restrict__ K,
                                                   const float* __restrict__ b,
                                                   float* __restrict__ ws) {
  __shared__ float red[9][8];
  float p[9];
#pragma unroll
  for (int t = 0; t < 9; ++t) p[t] = 0.0f;

  for (int i = threadIdx.x; i < 4096; i += 256) {
    const float* kp = K + i * 9;
#pragma unroll
    for (int t = 0; t < 9; ++t) p[t] += kp[t];
  }

  const int lane = threadIdx.x & 31;
  const int wave = threadIdx.x >> 5;

#pragma unroll
  for (int t = 0; t < 9; ++t) {
    float wsum;
#if __has_builtin(__builtin_amdgcn_wmma_f32_16x16x4_f32)
    // Layout-independent reduction: A = all-ones(16x4) => every row of D holds
    // the column sums of B, so sum(d[0] over 32 lanes) == 2 * sum(all B slots).
    v2f a; a.x = 1.0f; a.y = 1.0f;
    v2f bm; bm.x = p[t]; bm.y = 0.0f;
    v8f c = {0.f, 0.f, 0.f, 0.f, 0.f, 0.f, 0.f, 0.f};
    v8f d = __builtin_amdgcn_wmma_f32_16x16x4_f32(false, a, false, bm,
                                                  (short)0, c, false, false);
    float s = d[0];
#pragma unroll
    for (int off = 16; off; off >>= 1) s += __shfl_xor(s, off, 32);
    wsum = s * 0.5f;
#else
    float s = p[t];
#pragma unroll
    for (int off = 16; off; off >>= 1) s += __shfl_xor(s, off, 32);
    wsum = s;
#endif
    if (lane == 0) red[t][wave] = wsum;
  }
  __syncthreads();

  if (threadIdx.x == 0) {
#pragma unroll
    for (int t = 0; t < 9; ++t) {
      float s = 0.0f;
#pragma unroll
      for (int w = 0; w < 8; ++w) s += red[t][w];
      ws[t] = s;
    }
    ws[9] = b[0] * 36864.0f;  // 64*64*3*3
  }
}

// ---------- kernel 2: bandwidth-bound 3x3 depthwise stencil ----------
__global__ __launch_bounds__(256) void conv_main(const float* __restrict__ X,
                                                 const float* __restrict__ coef,
                                                 float* __restrict__ out) {
  __shared__ float tile[LROWS * STRD];  // 31552 B

  const int tid   = threadIdx.x;
  const int plane = blockIdx.x / BANDS;
  const int band  = blockIdx.x % BANDS;
  const int g0    = band * ROWS;

  const float* Xp = X   + (size_t)plane * (W * H);
  float*       Op = out + (size_t)plane * (W * H);

  // Zero the two halo columns actually consumed per row (x = -1 and x = 224).
  if (tid < 2 * LROWS) {
    int r = tid >> 1;
    int c = (tid & 1) ? 228 : 3;
    tile[r * STRD + c] = 0.0f;
  }

  // Fill 34 rows x 56 float4 of data: async global->LDS, or ds zero for pad rows.
  for (int s = tid; s < LROWS * 56; s += 256) {
    int r = s / 56, q = s % 56;
    int in_row = g0 - 1 + r;
    float* ldst = &tile[r * STRD + 4 + 4 * q];
    if ((unsigned)in_row < (unsigned)H) {
      async_load_f4(Xp + in_row * W + 4 * q, ldst);
    } else {
      v4f z = {0.f, 0.f, 0.f, 0.f};
      *(v4f*)ldst = z;
    }
  }
  wait_async0();
  __syncthreads();

  const float k00 = coef[0], k01 = coef[1], k02 = coef[2];
  const float k10 = coef[3], k11 = coef[4], k12 = coef[5];
  const float k20 = coef[6], k21 = coef[7], k22 = coef[8];
  const float bias = coef[9];

  for (int s = tid; s < ROWS * 56; s += 256) {  // exactly 7 iterations/thread
    int r = s / 56, q = s % 56;
    int x = 4 * q;  // output columns x .. x+3
    v4f acc = {bias, bias, bias, bias};
#pragma unroll
    for (int dy = 0; dy < 3; ++dy) {
      const float kl = (dy == 0) ? k00 : (dy == 1) ? k10 : k20;
      const float kc = (dy == 0) ? k01 : (dy == 1) ? k11 : k21;
      const float kr = (dy == 0) ? k02 : (dy == 1) ? k12 : k22;
      const float* row = &tile[(r + dy) * STRD + x];  // idx i holds input col (x-4+i)
      v4f A = *(const v4f*)(row);      // cols x-4 .. x-1
      v4f B = *(const v4f*)(row + 4);  // cols x   .. x+3
      v4f C = *(const v4f*)(row + 8);  // cols x+4 .. x+7
      // left tap (x-1..x+2)
      acc.x = fmaf(kl, A.w, acc.x);
      acc.y = fmaf(kl, B.x, acc.y);
      acc.z = fmaf(kl, B.y, acc.z);
      acc.w = fmaf(kl, B.z, acc.w);
      // center tap (x..x+3)
      acc.x = fmaf(kc, B.x, acc.x);
      acc.y = fmaf(kc, B.y, acc.y);
      acc.z = fmaf(kc, B.z, acc.z);
      acc.w = fmaf(kc, B.w, acc.w);
      // right tap (x+1..x+4)
      acc.x = fmaf(kr, B.y, acc.x);
      acc.y = fmaf(kr, B.z, acc.y);
      acc.z = fmaf(kr, B.w, acc.z);
      acc.w = fmaf(kr, C.x, acc.w);
    }
    // Streaming store: output is never re-read; keep L2 for cross-band halos.
    __builtin_nontemporal_store(acc, (v4f*)&Op[(g0 + r) * W + x]);
  }
}

extern "C" void kernel_launch(void* const* d_in, const int* in_sizes, int n_in,
                              void* d_out, int out_size, void* d_ws, size_t ws_size,
                              hipStream_t stream) {
  const float* X = (const float*)d_in[0];
  const float* K = (const float*)d_in[1];
  const float* b = (const float*)d_in[2];
  // d_in[3] = padding (==1), d_in[4] = stride (==1) per setup_inputs.
  float* ws = (float*)d_ws;  // ws[0..8] = Keff taps, ws[9] = bias_total

  conv_reduce<<<1, 256, 0, stream>>>(K, b, ws);
  conv_main<<<PLANES * BANDS, 256, 0, stream>>>(X, ws, (float*)d_out);
}